// UngroundedMicroProgram_3504693313604
// MI455X (gfx1250) — compile-verified
//
#include <hip/hip_runtime.h>
#include <stdint.h>

// Reference: S=524288 samples, record = (8 obj x 6 props) f32 = 192B.
// Needs props {0,1} of each object. d[j][q] = |x[s,0,q]-x[s,j+1,q]|,
// satisfies = any(d in [pb00,pb01]∩[pb10,pb11]) && exist-check,
// outputs: action_probs (2,S,3) then p_values (2,28,S) (4 duplicated d-planes).

#define EXIST_TH 0.8f

#define TPB    128                 // 4 waves (wave32)
#define SPT    2                   // samples per thread -> float2 pv stores
#define SPB    (TPB * SPT)         // 256 samples per block
#define RECB   192                 // bytes per record in global
#define NCHUNK (SPB * (RECB / 16)) // 3072 b128 chunks per block
#define CPT    (NCHUNK / TPB)      // 24 async loads per thread
#define LRECF  60                  // effective LDS record stride in floats (240B)
#define TILE_BYTES (SPB * LRECF * 4) // 61440 B dynamic LDS (<64KB)

typedef float v2f __attribute__((ext_vector_type(2)));

// LDS layout: 16B pad after every 4 chunks => la = 16*c + 16*(c>>2).
// Record r (12 chunks) starts at 240*r; shift-only math, constant increments.
__device__ __forceinline__ uint32_t lds_chunk_addr(uint32_t lbase, uint32_t c) {
    return lbase + (c << 4) + ((c >> 2) << 4);
}

__global__ __launch_bounds__(TPB)
void micro_program_kernel(const float* __restrict__ x,
                          const float* __restrict__ action,
                          const unsigned char* __restrict__ mask,
                          const float* __restrict__ pb,
                          float* __restrict__ out,
                          int S)
{
    extern __shared__ float tile[];
    const int tid       = threadIdx.x;
    const int blockBase = (int)blockIdx.x * SPB;
    const bool fullBlock = (blockBase + SPB) <= S;

    // ---- async global -> LDS staging (coalesced: lanes hit consecutive 16B) ----
    const uint64_t gbase = (uint64_t)(uintptr_t)x + (uint64_t)blockBase * RECB;
    const uint32_t lbase = (uint32_t)(uintptr_t)tile;   // low 32 bits = LDS byte addr

    uint32_t la = lds_chunk_addr(lbase, (uint32_t)tid);
    uint64_t ga = gbase + (uint64_t)tid * 16u;
    if (fullBlock) {
#pragma unroll
        for (int i = 0; i < CPT; ++i) {
            asm volatile("global_load_async_to_lds_b128 %0, %1, off"
                         :: "v"(la), "v"(ga) : "memory");
            la += (uint32_t)(TPB * 16 + TPB * 4);   // 2560: chunk + pad advance
            ga += (uint64_t)(TPB * 16);             // 2048
        }
    } else {
        const uint64_t glimit = (uint64_t)(uintptr_t)x + (uint64_t)S * RECB;
#pragma unroll
        for (int i = 0; i < CPT; ++i) {
            if (ga + 16u <= glimit) {
                asm volatile("global_load_async_to_lds_b128 %0, %1, off"
                             :: "v"(la), "v"(ga) : "memory");
            }
            la += (uint32_t)(TPB * 16 + TPB * 4);
            ga += (uint64_t)(TPB * 16);
        }
    }
    asm volatile("s_wait_asynccnt 0x0" ::: "memory");
    __syncthreads();

    // ---- uniform params (scalar loads) ----
    const float pb00 = pb[0], pb01 = pb[1], pb10 = pb[2], pb11 = pb[3];
    const float blo = fmaxf(pb00, pb10);     // d in [pb00,pb01] ∩ [pb10,pb11]
    const float bhi = fminf(pb01, pb11);     //   <=> d in [blo, bhi]
    const bool  m0 = (mask[0] != 0);
    const bool  m1 = (mask[1] != 0);
    const float an0 = action[0] / (action[0] + 1e-20f);
    const float an1 = action[1] / (action[1] + 1e-20f);
    const float an2 = action[2] / (action[2] + 1e-20f);

    const int sl0 = tid * SPT;
    const int s0  = blockBase + sl0;

    // padded float offsets of object o's (prop0,prop1) pair within a record
    constexpr int OFF[8] = {0, 6, 12, 22, 28, 34, 44, 50};

    float d[SPT][7][2];
    bool  sat[SPT];
#pragma unroll
    for (int k = 0; k < SPT; ++k) {
        const float* rec = tile + (sl0 + k) * LRECF;
        const v2f a = *(const v2f*)(rec + OFF[0]);          // obj0 props {0,1}
        bool enemyAll = true;
        bool any = false;
#pragma unroll
        for (int j = 0; j < 7; ++j) {
            const v2f b = *(const v2f*)(rec + OFF[j + 1]);  // obj j+1
            enemyAll = enemyAll && (b.y > EXIST_TH);
            const float d0 = fabsf(a.x - b.x);
            const float d1 = fabsf(a.y - b.y);
            d[k][j][0] = d0;
            d[k][j][1] = d1;
            any = any || (d0 >= blo && d0 <= bhi);
            any = any || (d1 >= blo && d1 <= bhi);
        }
        const bool exist = ((a.x > EXIST_TH) == m0) && (enemyAll == m1);
        sat[k] = any && exist;
    }

    // ---- unified store path (S is even, SPT=2: no half-pair tails exist) ----
    if (s0 + SPT <= S) {
        // action_probs (2,S,3): 6 consecutive floats per t -> 3 NT float2 stores
        const float u0 = sat[0] ? an0 : 0.0f, u1 = sat[0] ? an1 : 0.0f, u2 = sat[0] ? an2 : 0.0f;
        const float w0 = sat[1] ? an0 : 0.0f, w1 = sat[1] ? an1 : 0.0f, w2 = sat[1] ? an2 : 0.0f;
        const v2f p0 = {u0, u1}, p1 = {u2, w0}, p2 = {w1, w2};
#pragma unroll
        for (int t = 0; t < 2; ++t) {
            float* ap = out + (size_t)t * 3u * (size_t)S + (size_t)s0 * 3u;  // 8B aligned
            __builtin_nontemporal_store(p0, (v2f*)(ap + 0));
            __builtin_nontemporal_store(p1, (v2f*)(ap + 2));
            __builtin_nontemporal_store(p2, (v2f*)(ap + 4));
        }
        // p_values (2,28,S): plane (j*2+q)*2+r; t- and r-planes duplicate d
        float* pv = out + (size_t)6 * (size_t)S;
#pragma unroll
        for (int j = 0; j < 7; ++j) {
#pragma unroll
            for (int q = 0; q < 2; ++q) {
                const v2f v = {d[0][j][q], d[1][j][q]};
                const size_t o = (size_t)((j * 2 + q) * 2) * (size_t)S + (size_t)s0;
                __builtin_nontemporal_store(v, (v2f*)(pv + o));                    // t0,r0
                __builtin_nontemporal_store(v, (v2f*)(pv + o + (size_t)S));        // t0,r1
                __builtin_nontemporal_store(v, (v2f*)(pv + o + (size_t)28 * S));   // t1,r0
                __builtin_nontemporal_store(v, (v2f*)(pv + o + (size_t)29 * S));   // t1,r1
            }
        }
    }
}

extern "C" void kernel_launch(void* const* d_in, const int* in_sizes, int n_in,
                              void* d_out, int out_size, void* d_ws, size_t ws_size,
                              hipStream_t stream) {
    (void)n_in; (void)out_size; (void)d_ws; (void)ws_size;
    const float*         x      = (const float*)d_in[0];
    const float*         action = (const float*)d_in[1];
    const unsigned char* mask   = (const unsigned char*)d_in[2];
    const float*         pb     = (const float*)d_in[3];
    float*               out    = (float*)d_out;

    const int S = in_sizes[0] / 48;               // (S, 8, 6) float32
    const int blocks = (S + SPB - 1) / SPB;       // 2048 for S=524288
    hipLaunchKernelGGL(micro_program_kernel, dim3(blocks), dim3(TPB), TILE_BYTES, stream,
                       x, action, mask, pb, out, S);
}